// BlobRegressionLoss_82325933129960
// MI455X (gfx1250) — compile-verified
//
#include <hip/hip_runtime.h>

// ---------------------------------------------------------------------------
// TopK-BCE + SoftDice fused loss for gfx1250 (MI455X, wave32, WMMA)
//
// HBM-bound: 75.5 MB @ 23.3 TB/s ~ 3.2 us. Per-element math kept to a few
// VALU + 3 raw trans ops (v_exp_f32, v_log_f32, v_rcp_f32) so VALU stays
// under the memory roof. Top-k handled by histogram selection (no sort).
//
//   K0: zero count histogram in workspace
//   K1: fused streaming pass (float4 loads -> global_load_b128):
//        - stable BCE loss -> LDS count histogram (4096 bins over [0,16]),
//          flushed with deterministic u32 global atomics
//        - dice sums via V_WMMA_F32_16X16X4_F32 accumulators: diag of
//          p x t^T accumulates sum(p*t); B=ones gives row sums for
//          sum(p), sum(t); per-wave partials stored (no float atomics)
//   K2: finalize: suffix-scan histogram for top-k threshold, sum via bin
//       centers, fixed-order tree reduce of dice partials, emit scalar
// ---------------------------------------------------------------------------

#define BINS      4096
#define LMAX      16.0f
#define BIN_SCALE ((float)BINS / LMAX)     // 256.0
#define BIN_W     (LMAX / (float)BINS)     // 0.00390625
#define MAIN_B    256
#define MAIN_G    512
#define FIN_T     256

typedef float v2f __attribute__((ext_vector_type(2)));
typedef float v8f __attribute__((ext_vector_type(8)));

// stable BCE-with-logits + sigmoid using raw hardware trans ops.
//   e  = exp(-|x|) in (0,1]  -> 1+e in (1,2]: no denorm/inf/cancellation,
//   so raw v_log_f32 (log2) * ln2 and raw v_rcp_f32 are accurate to ~1e-7.
__device__ __forceinline__ void bce_elem(float x, float t, float& loss, float& p) {
    float ax  = __builtin_fabsf(x);
    float e   = __expf(-ax);                       // v_mul + v_exp_f32
    float op  = 1.0f + e;
    float lp  = __builtin_amdgcn_logf(op) * 0.6931471805599453f;  // v_log_f32 * ln2
    float sig = __builtin_amdgcn_rcpf(op);         // v_rcp_f32 = sigmoid(|x|)
    p    = (x >= 0.0f) ? sig : (1.0f - sig);
    loss = fmaxf(x, 0.0f) - x * t + lp;
}

__device__ __forceinline__ void hist_add(unsigned* lh, float loss) {
    int b = (int)(loss * BIN_SCALE);
    b = (b < 0) ? 0 : ((b > BINS - 1) ? (BINS - 1) : b);
    atomicAdd(&lh[b], 1u);
}

// select v[j] for runtime j in [0,7] (cndmask chain, no scratch)
__device__ __forceinline__ float v8_sel(v8f v, int j) {
    float r = v[0];
    r = (j == 1) ? v[1] : r;
    r = (j == 2) ? v[2] : r;
    r = (j == 3) ? v[3] : r;
    r = (j == 4) ? v[4] : r;
    r = (j == 5) ? v[5] : r;
    r = (j == 6) ? v[6] : r;
    r = (j == 7) ? v[7] : r;
    return r;
}

__global__ void zero_hist_kernel(unsigned* __restrict__ g_hist, int nwords) {
    int i = blockIdx.x * blockDim.x + threadIdx.x;
    if (i < nwords) g_hist[i] = 0u;
}

__global__ void __launch_bounds__(MAIN_B)
topk_dice_main_kernel(const float* __restrict__ logits,
                      const float* __restrict__ targets,
                      unsigned* __restrict__ g_hist,
                      float* __restrict__ g_part,   // [3 * nwaves]
                      int n, int nwaves) {
    __shared__ unsigned lh[BINS];
    for (int b = threadIdx.x; b < BINS; b += blockDim.x) lh[b] = 0u;
    __syncthreads();

    const float4* xl = (const float4*)logits;
    const float4* tl = (const float4*)targets;
    const int T      = gridDim.x * blockDim.x;
    const int gtid   = blockIdx.x * blockDim.x + threadIdx.x;
    const int nquads = n >> 2;
    const int full   = nquads / T;   // identical trip count -> EXEC all-1s at WMMA

    v8f c_pt = {0.f,0.f,0.f,0.f,0.f,0.f,0.f,0.f};
    v8f c_p  = {0.f,0.f,0.f,0.f,0.f,0.f,0.f,0.f};
    v8f c_t  = {0.f,0.f,0.f,0.f,0.f,0.f,0.f,0.f};
    v2f ones; ones[0] = 1.0f; ones[1] = 1.0f;
    float tpt = 0.f, tp = 0.f, tt = 0.f;   // scalar tail accumulators

    for (int it = 0; it < full; ++it) {
        int qidx = it * T + gtid;
        float4 xv = xl[qidx];             // global_load_b128
        float4 tv = tl[qidx];
        float l0, p0, l1, p1, l2, p2, l3, p3;
        bce_elem(xv.x, tv.x, l0, p0);
        bce_elem(xv.y, tv.y, l1, p1);
        bce_elem(xv.z, tv.z, l2, p2);
        bce_elem(xv.w, tv.w, l3, p3);
        hist_add(lh, l0); hist_add(lh, l1);
        hist_add(lh, l2); hist_add(lh, l3);
        v2f a0; a0[0] = p0;   a0[1] = p1;
        v2f b0; b0[0] = tv.x; b0[1] = tv.y;
        v2f a1; a1[0] = p2;   a1[1] = p3;
        v2f b1; b1[0] = tv.z; b1[1] = tv.w;
        // diag(c_pt) accumulates sum(p*t); B=ones rows accumulate sum(p)/sum(t)
        c_pt = __builtin_amdgcn_wmma_f32_16x16x4_f32(false, a0, false, b0,   (short)0, c_pt, false, false);
        c_p  = __builtin_amdgcn_wmma_f32_16x16x4_f32(false, a0, false, ones, (short)0, c_p,  false, false);
        c_t  = __builtin_amdgcn_wmma_f32_16x16x4_f32(false, b0, false, ones, (short)0, c_t,  false, false);
        c_pt = __builtin_amdgcn_wmma_f32_16x16x4_f32(false, a1, false, b1,   (short)0, c_pt, false, false);
        c_p  = __builtin_amdgcn_wmma_f32_16x16x4_f32(false, a1, false, ones, (short)0, c_p,  false, false);
        c_t  = __builtin_amdgcn_wmma_f32_16x16x4_f32(false, b1, false, ones, (short)0, c_t,  false, false);
    }

    // tail quad (scalar path, divergence allowed; no WMMA)
    {
        int qidx = full * T + gtid;
        if (qidx < nquads) {
            float4 xv = xl[qidx];
            float4 tv = tl[qidx];
            float l, p;
            bce_elem(xv.x, tv.x, l, p); hist_add(lh, l); tpt += p * tv.x; tp += p; tt += tv.x;
            bce_elem(xv.y, tv.y, l, p); hist_add(lh, l); tpt += p * tv.y; tp += p; tt += tv.y;
            bce_elem(xv.z, tv.z, l, p); hist_add(lh, l); tpt += p * tv.z; tp += p; tt += tv.z;
            bce_elem(xv.w, tv.w, l, p); hist_add(lh, l); tpt += p * tv.w; tp += p; tt += tv.w;
        }
    }
    // scalar remainder (n % 4 elements)
    {
        int ri = (nquads << 2) + gtid;
        if (ri < n) {
            float l, p;
            float xv = logits[ri], tv = targets[ri];
            bce_elem(xv, tv, l, p);
            hist_add(lh, l);
            tpt += p * tv; tp += p; tt += tv;
        }
    }
    __syncthreads();

    // flush LDS histogram with order-independent u32 atomics (deterministic)
    for (int b = threadIdx.x; b < BINS; b += blockDim.x) {
        unsigned v = lh[b];
        if (v) atomicAdd(&g_hist[b], v);
    }

    // ---- extract WMMA accumulators (16x16 f32 D layout: VGPR j holds
    //      row M=j at lanes 0-15 (N=lane) and row M=j+8 at lanes 16-31) ----
    const int lane = threadIdx.x & 31;
    // diagonal D[m][m]: m<8 -> VGPR m @ lane m ; m>=8 -> VGPR m-8 @ lane m+16
    int dj = (lane < 8) ? lane : (lane - 24);
    dj = (dj < 0) ? 0 : dj;
    float diag = v8_sel(c_pt, dj);
    float contrib_pt = (lane < 8 || lane >= 24) ? diag : 0.f;
    // column 0 of c_p / c_t: lane 0 holds rows 0..7, lane 16 holds rows 8..15
    float s8p = 0.f, s8t = 0.f;
#pragma unroll
    for (int j = 0; j < 8; ++j) { s8p += c_p[j]; s8t += c_t[j]; }
    float contrib_p = (lane == 0 || lane == 16) ? s8p : 0.f;
    float contrib_t = (lane == 0 || lane == 16) ? s8t : 0.f;
    contrib_pt += tpt; contrib_p += tp; contrib_t += tt;

    // wave32 shuffle reduce
    for (int off = 16; off > 0; off >>= 1) {
        contrib_pt += __shfl_xor(contrib_pt, off, 32);
        contrib_p  += __shfl_xor(contrib_p,  off, 32);
        contrib_t  += __shfl_xor(contrib_t,  off, 32);
    }
    if (lane == 0) {
        int wslot = blockIdx.x * (blockDim.x >> 5) + (threadIdx.x >> 5);
        g_part[wslot]              = contrib_pt;
        g_part[nwaves + wslot]     = contrib_p;
        g_part[2 * nwaves + wslot] = contrib_t;
    }
}

__global__ void __launch_bounds__(FIN_T)
finalize_kernel(const unsigned* __restrict__ g_hist,
                const float* __restrict__ g_part,
                int nwaves, unsigned K, float* __restrict__ out) {
    __shared__ float s_pt[FIN_T], s_p[FIN_T], s_t[FIN_T];
    __shared__ unsigned ccnt[FIN_T];
    __shared__ float csum[FIN_T];
    const int tid = threadIdx.x;

    // fixed-order reduction of dice partials (deterministic)
    const float* ppt = g_part;
    const float* pp  = g_part + nwaves;
    const float* pt_ = g_part + 2 * nwaves;
    float a = 0.f, b = 0.f, c = 0.f;
    for (int i = tid; i < nwaves; i += FIN_T) { a += ppt[i]; b += pp[i]; c += pt_[i]; }
    s_pt[tid] = a; s_p[tid] = b; s_t[tid] = c;

    // histogram chunk sums, chunk 0 = topmost 16 bins
    unsigned cc = 0; float cs = 0.f;
    int hi = BINS - 1 - tid * (BINS / FIN_T);
    for (int k = 0; k < BINS / FIN_T; ++k) {
        int bb = hi - k;
        unsigned nb = g_hist[bb];
        cc += nb;
        cs += (float)nb * (((float)bb + 0.5f) * BIN_W);
    }
    ccnt[tid] = cc; csum[tid] = cs;
    __syncthreads();

    for (int off = FIN_T / 2; off > 0; off >>= 1) {
        if (tid < off) {
            s_pt[tid] += s_pt[tid + off];
            s_p[tid]  += s_p[tid + off];
            s_t[tid]  += s_t[tid + off];
        }
        __syncthreads();
    }

    if (tid == 0) {
        // suffix scan: find the chunk, then the bin, where cumulative hits K
        unsigned long long cum = 0; float s = 0.f; int ci = 0;
        while (ci < FIN_T && cum + ccnt[ci] < (unsigned long long)K) {
            cum += ccnt[ci]; s += csum[ci]; ++ci;
        }
        float bce;
        if (ci >= FIN_T) {
            bce = (cum > 0) ? s / (float)cum : 0.f;   // degenerate guard
        } else {
            int hi2 = BINS - 1 - ci * (BINS / FIN_T);
            for (int k2 = 0; k2 < BINS / FIN_T; ++k2) {
                int bb = hi2 - k2;
                unsigned nb = g_hist[bb];
                float ctr = ((float)bb + 0.5f) * BIN_W;
                if (cum + nb >= (unsigned long long)K) {
                    unsigned r = K - (unsigned)cum;
                    s += (float)r * ctr;
                    break;
                }
                cum += nb; s += (float)nb * ctr;
            }
            bce = s / (float)K;
        }
        float I = s_pt[0];
        float U = s_p[0] + s_t[0];
        float dice = (2.0f * I + 1e-6f) / (U + 1e-6f);
        out[0] = bce + 0.5f * (1.0f - dice);
    }
}

extern "C" void kernel_launch(void* const* d_in, const int* in_sizes, int n_in,
                              void* d_out, int out_size, void* d_ws, size_t ws_size,
                              hipStream_t stream) {
    const float* logits  = (const float*)d_in[0];
    const float* targets = (const float*)d_in[1];
    const int n = in_sizes[0];                      // 9,437,184

    unsigned K = (unsigned)((double)n * 0.2);       // matches int(n*0.2)
    if (K < 1u) K = 1u;

    unsigned* g_hist = (unsigned*)d_ws;
    float*    g_part = (float*)((char*)d_ws + (size_t)BINS * sizeof(unsigned));
    const int nwaves = MAIN_G * (MAIN_B / 32);      // 4096 wave partials

    zero_hist_kernel<<<(BINS + 255) / 256, 256, 0, stream>>>(g_hist, BINS);
    topk_dice_main_kernel<<<MAIN_G, MAIN_B, 0, stream>>>(
        logits, targets, g_hist, g_part, n, nwaves);
    finalize_kernel<<<1, FIN_T, 0, stream>>>(g_hist, g_part, nwaves, K, (float*)d_out);
}